// GCNNDiagGaussianActor_33414845563487
// MI455X (gfx1250) — compile-verified
//
#include <hip/hip_runtime.h>
#include <hip/hip_bf16.h>
#include <math.h>

// ---------------------------------------------------------------------------
// GCN actor, algebraically collapsed:
//   fully-connected graph + self loops + uniform norm  ==>  per-group mean.
// Dense pipeline: G(1024x16) -> H1(1024x128) -> H2 -> M -> out(1024x4)
// GEMMs run on V_WMMA_F32_16X16X4_F32 (f32-exact, CDNA5 matrix pipe).
// Weights are pre-transposed so BOTH fragments load as contiguous b64.
// K and RELU are template params -> fully unrolled straight-line WMMA chain.
// ---------------------------------------------------------------------------

typedef __attribute__((ext_vector_type(2))) float v2f;
typedef __attribute__((ext_vector_type(8))) float v8f;

#define BS        1024
#define NODES     32
#define OBSDIM    16
#define FEAT      14
#define HID       128
#define OUT4      4

// ---- Phase 0: per-batch mean of node features, padded K 14 -> 16 ----------
__global__ __launch_bounds__(256) void k_reduce_obs(const float* __restrict__ obs,
                                                    float* __restrict__ G) {
    int i = blockIdx.x * blockDim.x + threadIdx.x;   // 0 .. 1024*16-1
    if (i >= BS * 16) return;
    int b = i >> 4;
    int k = i & 15;
    float s = 0.0f;
    if (k < FEAT) {
        const float* p = obs + b * (NODES * OBSDIM) + 2 + k;  // skip ACT_DIM cols
        #pragma unroll
        for (int n = 0; n < NODES; ++n) s += p[n * OBSDIM];
        s *= (1.0f / 32.0f);
    }
    G[i] = s;
}

// ---- Phase 0b: pad+transpose W1 (14x128) -> W1t (128x16), pad rows zero ---
__global__ __launch_bounds__(256) void k_prep_w1t(const float* __restrict__ W1,
                                                  float* __restrict__ W1t) {
    int i = blockIdx.x * blockDim.x + threadIdx.x;   // 0 .. 128*16-1
    if (i >= HID * 16) return;
    int c = i >> 4;          // output column (N)
    int r = i & 15;          // K index
    W1t[(size_t)c * 16 + r] = (r < FEAT) ? W1[r * HID + c] : 0.0f;
}

// ---- Phase 0c: transpose 128x128 weight -> column-major -------------------
__global__ __launch_bounds__(256) void k_transpose128(const float* __restrict__ W,
                                                      float* __restrict__ Wt) {
    int i = blockIdx.x * blockDim.x + threadIdx.x;   // 0 .. 128*128-1
    if (i >= HID * HID) return;
    int c = i >> 7;          // output column (N)
    int r = i & 127;         // K index
    Wt[(size_t)c * HID + r] = W[(size_t)r * HID + c];
}

// ---- WMMA f32 GEMM: C(1024xHID) = relu?(A(1024xK) @ B(KxHID) + bias) ------
// B passed TRANSPOSED (HID x K). One wave per 16x16 C tile, 4 waves/block.
// A frag (16x4 f32): lane L -> row M=L%16, a.x=K(2*(L/16)), a.y=K(2*(L/16)+1)
// B frag (4x16 f32): mirrored: VGPR0 lanes0-15 K=0 / lanes16-31 K=2, VGPR1 K=1/K=3
//   -> with Bt, lane L reads Bt[(col_base+L%16)*K + 2*(L/16)] as one b64.
// C/D (16x16 f32):   VGPR r, lanes0-15 -> M=r, lanes16-31 -> M=r+8, N=L%16
template <int K, bool RELU>
__global__ __launch_bounds__(128) void k_gemm_wmma(const float* __restrict__ A,
                                                   const float* __restrict__ Bt,
                                                   const float* __restrict__ bias,
                                                   float* __restrict__ C) {
    const int lane = threadIdx.x;                    // 0..31
    const int half = lane >> 4;                      // 0 or 1
    const int l16  = lane & 15;
    const int col_base = (blockIdx.x * 4 + threadIdx.y) * 16;
    const int row_base = blockIdx.y * 16;

    // bias depends only on column -> same value in all 8 accumulator VGPRs
    const float bv = bias[col_base + l16];
    v8f c = {bv, bv, bv, bv, bv, bv, bv, bv};

    const float* __restrict__ Arow = A  + (size_t)(row_base + l16) * K + 2 * half;
    const float* __restrict__ Brow = Bt + (size_t)(col_base + l16) * K + 2 * half;

    #pragma unroll
    for (int k0 = 0; k0 < K; k0 += 4) {
        v2f a = *(const v2f*)(Arow + k0);            // global_load_b64
        v2f b = *(const v2f*)(Brow + k0);            // global_load_b64
        // D = A x B + C   (emits v_wmma_f32_16x16x4_f32)
        c = __builtin_amdgcn_wmma_f32_16x16x4_f32(
                /*neg_a=*/false, a, /*neg_b=*/false, b,
                /*c_mod=*/(short)0, c, /*reuse_a=*/false, /*reuse_b=*/false);
    }

    const int crow = row_base + half * 8;
    #pragma unroll
    for (int r = 0; r < 8; ++r) {
        float v = c[r];
        if (RELU) v = fmaxf(v, 0.0f);
        C[(size_t)(crow + r) * HID + col_base + l16] = v;
    }
}

// ---- Head: out = M @ Wm2 + bm2 ; mu / std ; broadcast x32 per batch -------
__global__ __launch_bounds__(64) void k_head(const float* __restrict__ Mm,
                                             const float* __restrict__ Wm2,
                                             const float* __restrict__ bm2,
                                             float* __restrict__ out) {
    const int b = blockIdx.x;      // batch
    const int t = threadIdx.x;     // 0..63
    __shared__ float sm[HID];
    __shared__ float res[OUT4];

    sm[t]      = Mm[(size_t)b * HID + t];
    sm[t + 64] = Mm[(size_t)b * HID + t + 64];
    __syncthreads();

    if (t < OUT4) {
        float acc = bm2[t];
        #pragma unroll 8
        for (int k = 0; k < HID; ++k) acc += sm[k] * Wm2[k * OUT4 + t];
        res[t] = acc;
    }
    __syncthreads();

    const float mu0 = res[0], mu1 = res[1];
    // log_std = -5 + 3.5*(tanh(x)+1) ; std = exp(log_std)
    const float s0 = __expf(-5.0f + 3.5f * (tanhf(res[2]) + 1.0f));
    const float s1 = __expf(-5.0f + 3.5f * (tanhf(res[3]) + 1.0f));

    const float mu = (t & 1) ? mu1 : mu0;
    const float sd = (t & 1) ? s1 : s0;
    // output shape (2, 1024, 64) flat
    out[(size_t)b * 64 + t]                   = mu;
    out[(size_t)BS * 64 + (size_t)b * 64 + t] = sd;
}

extern "C" void kernel_launch(void* const* d_in, const int* in_sizes, int n_in,
                              void* d_out, int out_size, void* d_ws, size_t ws_size,
                              hipStream_t stream) {
    const float* obs = (const float*)d_in[0];   // (1024, 512)
    const float* W1  = (const float*)d_in[1];   // (14, 128)
    const float* b1  = (const float*)d_in[2];   // (128,)
    const float* W2  = (const float*)d_in[3];   // (128, 128)
    const float* b2  = (const float*)d_in[4];   // (128,)
    const float* Wm1 = (const float*)d_in[5];   // (128, 128)
    const float* bm1 = (const float*)d_in[6];   // (128,)
    const float* Wm2 = (const float*)d_in[7];   // (128, 4)
    const float* bm2 = (const float*)d_in[8];   // (4,)
    float* out = (float*)d_out;                 // (2, 1024, 64)

    // workspace layout (floats)
    float* ws   = (float*)d_ws;
    float* G    = ws;                       // 1024*16  = 16384
    float* W1t  = G    + BS * 16;           // 128*16   = 2048
    float* W2t  = W1t  + HID * 16;          // 128*128  = 16384
    float* Wm1t = W2t  + HID * HID;         // 128*128  = 16384
    float* H1   = Wm1t + HID * HID;         // 1024*128 = 131072
    float* H2   = H1   + BS * HID;          // 1024*128
    float* Mm   = H2   + BS * HID;          // 1024*128
    (void)ws_size; (void)n_in; (void)in_sizes; (void)out_size;

    // Phase 0: collapse graph aggregation to per-group mean; prep weights
    k_reduce_obs <<<(BS * 16 + 255) / 256, 256, 0, stream>>>(obs, G);
    k_prep_w1t   <<<(HID * 16 + 255) / 256, 256, 0, stream>>>(W1, W1t);
    k_transpose128<<<(HID * HID + 255) / 256, 256, 0, stream>>>(W2, W2t);
    k_transpose128<<<(HID * HID + 255) / 256, 256, 0, stream>>>(Wm1, Wm1t);

    // Phase 1-3: dense GEMMs on the WMMA pipe, bias + ReLU fused.
    // grid: 2 x 64 blocks, 4 waves/block, one 16x16 C tile per wave.
    dim3 grid(HID / 16 / 4, BS / 16);
    dim3 blk(32, 4);
    k_gemm_wmma<16,  true><<<grid, blk, 0, stream>>>(G,  W1t,  b1,  H1);
    k_gemm_wmma<HID, true><<<grid, blk, 0, stream>>>(H1, W2t,  b2,  H2);
    k_gemm_wmma<HID, true><<<grid, blk, 0, stream>>>(H2, Wm1t, bm1, Mm);

    // Phase 4: tiny head (128-dots, tanh/exp), broadcast x32 per batch
    k_head<<<BS, 64, 0, stream>>>(Mm, Wm2, bm2, out);
}